// LTMModule_31258771980626
// MI455X (gfx1250) — compile-verified
//
#include <hip/hip_runtime.h>

typedef __attribute__((ext_vector_type(2))) float v2f;
typedef __attribute__((ext_vector_type(8))) float v8f;

#define N_SLOTS   1024
#define KEY_DIM   64
#define VAL_DIM   64
#define TOPK      4
#define LR_C      0.001f
#define MOM_C     0.9f
#define WD_C      0.0001f

#define WAVE      32
#define WPB       4                 // waves per block
#define THREADS   (WPB * WAVE)
#define M_TILES   4                 // 16-query tiles per wave -> 64 queries/wave

#define WMMA_F32(A, B, C) \
    __builtin_amdgcn_wmma_f32_16x16x4_f32(false, (A), false, (B), (short)0, (C), false, false)

// Wave-private LDS ordering: CDNA5 LDS ops from one wave complete in order,
// so cross-lane store->load within a wave needs only a compiler fence.
#define WAVE_SYNC() do { __builtin_amdgcn_wave_barrier(); asm volatile("" ::: "memory"); } while (0)

// ---------------------------------------------------------------------------
// Kernel 0: zero the scatter workspace (slot_grads[1024*64] ++ counts[1024])
// ---------------------------------------------------------------------------
__global__ void ltm_zero_ws(float* __restrict__ ws, int n) {
    int i = blockIdx.x * blockDim.x + threadIdx.x;
    if (i < n) ws[i] = 0.0f;
}

// Branchless sorted top-4 insert (descending; strict > keeps the lowest index
// on ties, matching jax.lax.top_k).
__device__ __forceinline__ void top4_insert(float v, int gi, float tv[4], int ti[4]) {
    const bool gt0 = v > tv[0];
    const bool gt1 = v > tv[1];
    const bool gt2 = v > tv[2];
    const bool gt3 = v > tv[3];
    float nv3 = gt2 ? tv[2] : (gt3 ? v : tv[3]);
    int   ni3 = gt2 ? ti[2] : (gt3 ? gi : ti[3]);
    float nv2 = gt1 ? tv[1] : (gt2 ? v : tv[2]);
    int   ni2 = gt1 ? ti[1] : (gt2 ? gi : ti[2]);
    float nv1 = gt0 ? tv[0] : (gt1 ? v : tv[1]);
    int   ni1 = gt0 ? ti[0] : (gt1 ? gi : ti[1]);
    float nv0 = gt0 ? v  : tv[0];
    int   ni0 = gt0 ? gi : ti[0];
    tv[0] = nv0; ti[0] = ni0;
    tv[1] = nv1; ti[1] = ni1;
    tv[2] = nv2; ti[2] = ni2;
    tv[3] = nv3; ti[3] = ni3;
}

// ---------------------------------------------------------------------------
// Kernel 1: fused sim-GEMM (WMMA f32 16x16x4, 4-way M-blocked) + top-4 +
//           gather + grad scatter. One wave32 owns 64 queries.
// ---------------------------------------------------------------------------
__global__ __launch_bounds__(THREADS) void ltm_main(
    const float* __restrict__ queries,   // [Nq, 64]
    const float* __restrict__ keys,      // [1024, 64]
    const float* __restrict__ vals,      // [1024, 64]
    const float* __restrict__ grads,     // [Nq, 4, 64]
    float* __restrict__ slot_grads,      // ws [1024, 64]
    float* __restrict__ counts,          // ws [1024]
    float* __restrict__ retrieved)       // out [Nq, 4, 64]
{
    __shared__ float simTile[WPB][2][16][17];   // two wave-private staging bufs
    __shared__ int   idxBuf[WPB][16 * M_TILES][TOPK];

    const int lane  = threadIdx.x & (WAVE - 1);
    const int w     = threadIdx.x >> 5;
    const int wave  = blockIdx.x * WPB + w;
    const int qBase = wave * (16 * M_TILES);

    // WMMA f32 16x16x4 operand layout (ISA 7.12.2):
    //  A (16x4): lanes 0-15 row M=lane hold K=0,1; lanes 16-31 hold K=2,3.
    //  B (4x16): lanes 0-15 col N=lane hold K=0,1; lanes 16-31 hold K=2,3.
    // Both map to a contiguous 8-byte load from row-major [row][k0+koff].
    const int row  = lane & 15;
    const int koff = (lane >> 4) * 2;

    // Preload A fragments for all 4 M-tiles (64 queries x 64 dims, in regs).
    v2f a0[16], a1[16], a2[16], a3[16];
    {
        const float* q0 = queries + (size_t)(qBase + row) * KEY_DIM + koff;
        #pragma unroll
        for (int kk = 0; kk < 16; ++kk) {
            a0[kk] = *(const v2f*)(q0 + kk * 4);
            a1[kk] = *(const v2f*)(q0 + 16 * KEY_DIM + kk * 4);
            a2[kk] = *(const v2f*)(q0 + 32 * KEY_DIM + kk * 4);
            a3[kk] = *(const v2f*)(q0 + 48 * KEY_DIM + kk * 4);
        }
    }

    // Two top-4 sets per lane:
    //   lanes 0-15  own (tile0,row) via set0 and (tile1,row) via set1
    //   lanes 16-31 own (tile2,row) via set0 and (tile3,row) via set1
    float tv0[4] = {-3.0e38f, -3.0e38f, -3.0e38f, -3.0e38f};
    float tv1[4] = {-3.0e38f, -3.0e38f, -3.0e38f, -3.0e38f};
    int   ti0[4] = {0, 0, 0, 0};
    int   ti1[4] = {0, 0, 0, 0};

    const int half = lane >> 4;          // which staging buffer this lane reads
    const int mHi  = half * 8;           // C layout: VGPR r -> rows r / r+8

    for (int nt = 0; nt < N_SLOTS / 16; ++nt) {
        const int slotBase = nt * 16;

        // Batch-load all 16 B fragments for this slot tile (single clause).
        v2f b[16];
        const float* krow = keys + (size_t)(slotBase + row) * KEY_DIM + koff;
        #pragma unroll
        for (int kk = 0; kk < 16; ++kk)
            b[kk] = *(const v2f*)(krow + kk * 4);

        // 64 WMMAs, 4 independent accumulator chains sharing each B fragment.
        v8f c0 = {}, c1 = {}, c2 = {}, c3 = {};
        #pragma unroll
        for (int kk = 0; kk < 16; ++kk) {
            c0 = WMMA_F32(a0[kk], b[kk], c0);
            c1 = WMMA_F32(a1[kk], b[kk], c1);
            c2 = WMMA_F32(a2[kk], b[kk], c2);
            c3 = WMMA_F32(a3[kk], b[kk], c3);
        }

        // Merge round 0: tiles 0 (buf0) and 2 (buf1). Wave-private staging,
        // so only wave-level ordering fences are required (no s_barrier).
        #pragma unroll
        for (int r = 0; r < 8; ++r) {
            simTile[w][0][r + mHi][lane & 15] = c0[r];
            simTile[w][1][r + mHi][lane & 15] = c2[r];
        }
        WAVE_SYNC();
        #pragma unroll
        for (int j = 0; j < 16; ++j)
            top4_insert(simTile[w][half][row][j], slotBase + j, tv0, ti0);
        WAVE_SYNC();

        // Merge round 1: tiles 1 (buf0) and 3 (buf1).
        #pragma unroll
        for (int r = 0; r < 8; ++r) {
            simTile[w][0][r + mHi][lane & 15] = c1[r];
            simTile[w][1][r + mHi][lane & 15] = c3[r];
        }
        WAVE_SYNC();
        #pragma unroll
        for (int j = 0; j < 16; ++j)
            top4_insert(simTile[w][half][row][j], slotBase + j, tv1, ti1);
        WAVE_SYNC();
    }

    // Publish indices: lane (half,row) owns query rows t0*16+row and (t0+1)*16+row.
    {
        const int t0 = half * 2;
        #pragma unroll
        for (int j = 0; j < TOPK; ++j) {
            idxBuf[w][t0 * 16 + row][j]       = ti0[j];
            idxBuf[w][(t0 + 1) * 16 + row][j] = ti1[j];
        }
    }
    WAVE_SYNC();

    // Gather retrieved = vals[idx] and scatter grads into slot accumulators.
    // 64 dims / 32 lanes -> 2 floats per lane.
    const int d = lane * 2;
    for (int m = 0; m < 16 * M_TILES; ++m) {
        const size_t q = (size_t)(qBase + m);
        __builtin_prefetch(grads + (q * TOPK) * VAL_DIM + 1024, 0, 1);
        #pragma unroll
        for (int j = 0; j < TOPK; ++j) {
            const int slot = idxBuf[w][m][j];

            v2f vv = *(const v2f*)(vals + (size_t)slot * VAL_DIM + d);
            *(v2f*)(retrieved + (q * TOPK + j) * VAL_DIM + d) = vv;

            v2f g = *(const v2f*)(grads + (q * TOPK + j) * VAL_DIM + d);
            atomicAdd(&slot_grads[slot * VAL_DIM + d],     g.x);
            atomicAdd(&slot_grads[slot * VAL_DIM + d + 1], g.y);
            if (lane == 0) atomicAdd(&counts[slot], 1.0f);
        }
    }
}

// ---------------------------------------------------------------------------
// Kernel 2: mean + momentum + weight-decay update on [1024, 64]
// ---------------------------------------------------------------------------
__global__ void ltm_finalize(
    const float* __restrict__ vals,
    const float* __restrict__ mom,
    const float* __restrict__ slot_grads,
    const float* __restrict__ counts,
    float* __restrict__ vals_new,
    float* __restrict__ mom_new)
{
    int i = blockIdx.x * blockDim.x + threadIdx.x;   // 1024*64 threads
    int slot = i >> 6;
    float cnt = counts[slot];
    bool  nz  = cnt > 0.0f;
    float sg  = slot_grads[i];
    if (nz) sg /= fmaxf(cnt, 1.0f);
    float m = MOM_C * mom[i] + sg;
    float v = vals[i];
    float delta = -LR_C * (m + WD_C * v);
    vals_new[i] = v + (nz ? delta : 0.0f);
    mom_new[i]  = m;
}

// ---------------------------------------------------------------------------
extern "C" void kernel_launch(void* const* d_in, const int* in_sizes, int n_in,
                              void* d_out, int out_size, void* d_ws, size_t ws_size,
                              hipStream_t stream) {
    const float* queries = (const float*)d_in[0];
    const float* keys    = (const float*)d_in[1];
    const float* vals    = (const float*)d_in[2];
    const float* mom     = (const float*)d_in[3];
    const float* grads   = (const float*)d_in[4];

    const int nQ = in_sizes[0] / KEY_DIM;            // 262144

    float* out       = (float*)d_out;
    float* retrieved = out;                                        // [nQ,4,64]
    float* vals_new  = out + (size_t)nQ * TOPK * VAL_DIM;          // [1024,64]
    float* mom_new   = vals_new + N_SLOTS * VAL_DIM;               // [1024,64]

    float* slot_grads = (float*)d_ws;                              // [1024,64]
    float* cnts       = slot_grads + N_SLOTS * VAL_DIM;            // [1024]

    const int zn = N_SLOTS * VAL_DIM + N_SLOTS;
    ltm_zero_ws<<<(zn + 255) / 256, 256, 0, stream>>>(slot_grads, zn);

    const int nWaves = nQ / (16 * M_TILES);          // 4096 waves
    ltm_main<<<nWaves / WPB, THREADS, 0, stream>>>(
        queries, keys, vals, grads, slot_grads, cnts, retrieved);

    ltm_finalize<<<(N_SLOTS * VAL_DIM) / 256, 256, 0, stream>>>(
        vals, mom, slot_grads, cnts, vals_new, mom_new);
}